// DPLRSSMBlock_60138132079079
// MI455X (gfx1250) — compile-verified
//
#include <hip/hip_runtime.h>

// DPLR SSM block for MI455X (gfx1250, wave32, WMMA).
//   A = diag(a) + p q^T            (H=2048, R=1)
//   expA = expm(delta*A)           -> scaling(2^-5) + 10-term Taylor (Horner), all GEMMs
//   A^{-1} via Sherman-Morrison    -> O(H^2), no SVD
//   B_disc = A^{-1}(expA - I) b    -> rank-1-corrected diag scale + 1 GEMM
//   out = h @ expA^T + x @ B_disc^T
// All GEMMs run on V_WMMA_F32_16X16X4_F32 (fp32 matrix cores).
// LDS tiles are stored K-pair-interleaved so each WMMA operand fragment is a
// single aligned ds_load_b64 (no VGPR repacking). Global->LDS staging is
// register double-buffered so VMEM latency overlaps the WMMA stream.

typedef __attribute__((ext_vector_type(2))) float v2f;
typedef __attribute__((ext_vector_type(8))) float v8f;

#define HDIM 2048
static_assert((HDIM % 128) == 0, "tile divisibility");

constexpr int TAYLOR_N  = 10;  // Taylor order
constexpr int SQUARINGS = 5;   // scaling s: X = delta*A * 2^-s

// ---------------------------------------------------------------------------
// Elementwise / vector helper kernels (all O(H^2) or O(H))
// ---------------------------------------------------------------------------

// X[i][j] = scale * delta * ( p[i]*q[j] + (i==j)*a[i] )
__global__ __launch_bounds__(256) void build_x_kernel(
    const float* __restrict__ a_diag, const float* __restrict__ p,
    const float* __restrict__ q, const float* __restrict__ delta,
    float scale, float* __restrict__ X)
{
  size_t idx = (size_t)blockIdx.x * 256 + threadIdx.x;
  int i = (int)(idx >> 11);
  int j = (int)(idx & (HDIM - 1));
  float v = p[i] * q[j];
  if (i == j) v += a_diag[i];
  X[idx] = scale * delta[0] * v;
}

// R = I + X * invn   (Horner seed)
__global__ __launch_bounds__(256) void r_init_kernel(
    const float* __restrict__ X, float* __restrict__ R, float invn)
{
  size_t idx = (size_t)blockIdx.x * 256 + threadIdx.x;
  int i = (int)(idx >> 11);
  int j = (int)(idx & (HDIM - 1));
  R[idx] = X[idx] * invn + ((i == j) ? 1.0f : 0.0f);
}

// u = p/a, v = q/a, denom = 1 + sum(q*p/a)   (Sherman-Morrison pieces)
__global__ __launch_bounds__(256) void uvd_kernel(
    const float* __restrict__ a, const float* __restrict__ p,
    const float* __restrict__ q, float* __restrict__ u,
    float* __restrict__ v, float* __restrict__ dn)
{
  __shared__ float red[256];
  float s = 0.0f;
  for (int i = threadIdx.x; i < HDIM; i += 256) {
    float ai = a[i];
    float ui = p[i] / ai;
    u[i] = ui;
    v[i] = q[i] / ai;
    s += q[i] * ui;
  }
  red[threadIdx.x] = s;
  __syncthreads();
  for (int off = 128; off > 0; off >>= 1) {
    if ((int)threadIdx.x < off) red[threadIdx.x] += red[threadIdx.x + off];
    __syncthreads();
  }
  if (threadIdx.x == 0) dn[0] = 1.0f + red[0];
}

// w[j] = sum_i v[i] * (E[i][j] - (i==j)) = (v^T (E - I))[j]
__global__ __launch_bounds__(256) void w_kernel(
    const float* __restrict__ v, const float* __restrict__ E,
    float* __restrict__ w)
{
  int j = blockIdx.x * 256 + threadIdx.x;
  float s = 0.0f;
  for (int i = 0; i < HDIM; ++i) s += v[i] * E[(size_t)i * HDIM + j];
  w[j] = s - v[j];
}

// C1 = A^{-1} (E - I) = D^{-1}(E-I) - u w^T / denom
__global__ __launch_bounds__(256) void c1_kernel(
    const float* __restrict__ E, const float* __restrict__ a,
    const float* __restrict__ u, const float* __restrict__ w,
    const float* __restrict__ dn, float* __restrict__ C1)
{
  size_t idx = (size_t)blockIdx.x * 256 + threadIdx.x;
  int i = (int)(idx >> 11);
  int j = (int)(idx & (HDIM - 1));
  float m = E[idx] - ((i == j) ? 1.0f : 0.0f);
  C1[idx] = m / a[i] - u[i] * w[j] / dn[0];
}

// ---------------------------------------------------------------------------
// FP32 WMMA GEMM:  C = alpha * A @ op(B)  [+ C_in] [+ I]
//   A: [M][K] row-major.  B: NN -> [K][N], NT -> [N][K] (op(B)=B^T), row-major.
//   Block = 256 threads = 8 waves; 128x128 tile; wave tile 32x64 (2x4 WMMA accs).
//   LDS layout: K-pair-interleaved v2f elements -> one ds_load_b64 per fragment.
//   Row pitch 144 v2f = 288 words == 32 banks (mod 64): the two lane-halves
//   (reading kp and kp+1) hit disjoint bank halves -> conflict-free.
// ---------------------------------------------------------------------------
template <bool TRANSB, bool ADD_I, bool ACCUM>
__global__ __launch_bounds__(256) void gemm_wmma_f32(
    const float* __restrict__ A, const float* __restrict__ Bm,
    float* __restrict__ C, int M, int N, int K, float alpha)
{
  constexpr int TM = 128, TN = 128, TK = 16;
  constexpr int APITCH = TM + 16;   // v2f elements per kp-row
  constexpr int BPITCH = TN + 16;
  __shared__ v2f As2[TK / 2][APITCH];  // As2[kp][m] = {A[m][2kp], A[m][2kp+1]}
  __shared__ v2f Bs2[TK / 2][BPITCH];  // Bs2[kp][n] = {opB[2kp][n], opB[2kp+1][n]}

  const int tid  = threadIdx.x;
  const int lane = tid & 31;
  const int l16  = lane & 15;
  const int half = lane >> 4;       // lane half selects K pair / C row half
  const int wave = tid >> 5;
  const int wr   = (wave & 3) * 32; // 4 waves down
  const int wc   = (wave >> 2) * 64;// 2 waves across
  const int br   = blockIdx.y * TM;
  const int bc   = blockIdx.x * TN;

  // Per-thread staging coordinates (A and B-NT: float4 along K; B-NN: along N)
  const int aM[2]  = { (tid + 0) >> 2, (tid + 256) >> 2 };     // 0..127
  const int aK4    = (tid & 3) << 2;                           // 0,4,8,12
  const int bKp    = tid >> 5;                                 // 0..7   (NN)
  const int bN4    = (tid & 31) << 2;                          // 0..124 (NN)

  const v8f vzero = {0.f, 0.f, 0.f, 0.f, 0.f, 0.f, 0.f, 0.f};
  v8f acc[2][4];
#pragma unroll
  for (int mt = 0; mt < 2; ++mt)
#pragma unroll
    for (int nt = 0; nt < 4; ++nt) acc[mt][nt] = vzero;

  float4 aReg[2], bReg[2];

  auto load_global = [&](int kb) {
#pragma unroll
    for (int r = 0; r < 2; ++r)
      aReg[r] = *(const float4*)(A + (size_t)(br + aM[r]) * K + kb + aK4);
    if (!TRANSB) {
      bReg[0] = *(const float4*)(Bm + (size_t)(kb + 2 * bKp + 0) * N + bc + bN4);
      bReg[1] = *(const float4*)(Bm + (size_t)(kb + 2 * bKp + 1) * N + bc + bN4);
    } else {
#pragma unroll
      for (int r = 0; r < 2; ++r)
        bReg[r] = *(const float4*)(Bm + (size_t)(bc + aM[r]) * K + kb + aK4);
    }
  };

  auto store_lds = [&]() {
#pragma unroll
    for (int r = 0; r < 2; ++r) {
      v2f lo; lo[0] = aReg[r].x; lo[1] = aReg[r].y;
      v2f hi; hi[0] = aReg[r].z; hi[1] = aReg[r].w;
      As2[(aK4 >> 1) + 0][aM[r]] = lo;
      As2[(aK4 >> 1) + 1][aM[r]] = hi;
    }
    if (!TRANSB) {
      const float* b0 = &bReg[0].x;
      const float* b1 = &bReg[1].x;
#pragma unroll
      for (int i = 0; i < 4; ++i) {
        v2f e; e[0] = b0[i]; e[1] = b1[i];
        Bs2[bKp][bN4 + i] = e;
      }
    } else {
#pragma unroll
      for (int r = 0; r < 2; ++r) {
        v2f lo; lo[0] = bReg[r].x; lo[1] = bReg[r].y;
        v2f hi; hi[0] = bReg[r].z; hi[1] = bReg[r].w;
        Bs2[(aK4 >> 1) + 0][aM[r]] = lo;
        Bs2[(aK4 >> 1) + 1][aM[r]] = hi;
      }
    }
  };

  load_global(0);  // prologue

  for (int kb = 0; kb < K; kb += TK) {
    store_lds();
    __syncthreads();

    if (kb + TK < K) load_global(kb + TK);  // overlap VMEM with WMMA stream

    // ---- 4 k-steps of V_WMMA_F32_16X16X4_F32 per stage ----
#pragma unroll
    for (int t = 0; t < 4; ++t) {
      const int kp = t * 2 + half;  // ISA A/B layout: lanes16-31 hold K+2
      v2f af[2];
      v2f bf[4];
#pragma unroll
      for (int mt = 0; mt < 2; ++mt) af[mt] = As2[kp][wr + mt * 16 + l16];
#pragma unroll
      for (int nt = 0; nt < 4; ++nt) bf[nt] = Bs2[kp][wc + nt * 16 + l16];
#pragma unroll
      for (int mt = 0; mt < 2; ++mt)
#pragma unroll
        for (int nt = 0; nt < 4; ++nt)
          acc[mt][nt] = __builtin_amdgcn_wmma_f32_16x16x4_f32(
              false, af[mt], false, bf[nt], (short)0, acc[mt][nt], false, false);
    }
    __syncthreads();
  }

  // ---- writeback: C VGPR r holds M=r (lanes 0-15) / M=r+8 (lanes 16-31) ----
#pragma unroll
  for (int mt = 0; mt < 2; ++mt)
#pragma unroll
    for (int nt = 0; nt < 4; ++nt)
#pragma unroll
      for (int r = 0; r < 8; ++r) {
        int row = br + wr + mt * 16 + half * 8 + r;
        int col = bc + wc + nt * 16 + l16;
        size_t o = (size_t)row * N + col;
        float v = alpha * acc[mt][nt][r];
        if (ACCUM) v += C[o];
        if (ADD_I) { if (row == col) v += 1.0f; }
        C[o] = v;
      }
}

// ---------------------------------------------------------------------------
// Host-side orchestration (all launches on `stream`; deterministic)
// ---------------------------------------------------------------------------
extern "C" void kernel_launch(void* const* d_in, const int* in_sizes, int n_in,
                              void* d_out, int out_size, void* d_ws, size_t ws_size,
                              hipStream_t stream) {
  (void)in_sizes; (void)n_in; (void)out_size; (void)ws_size;

  const float* h      = (const float*)d_in[0];
  const float* x      = (const float*)d_in[1];
  const float* a_diag = (const float*)d_in[2];
  const float* p_vec  = (const float*)d_in[3];
  const float* q_vec  = (const float*)d_in[4];
  const float* b_mat  = (const float*)d_in[5];
  const float* delta  = (const float*)d_in[6];
  float* out = (float*)d_out;

  const size_t MATB = (size_t)HDIM * HDIM * sizeof(float);  // 16 MB
  char* ws  = (char*)d_ws;
  float* M0 = (float*)(ws);             // X, later reused for C1
  float* M1 = (float*)(ws + MATB);      // expm ping
  float* M2 = (float*)(ws + 2 * MATB);  // expm pong
  float* u  = (float*)(ws + 3 * MATB);
  float* v  = u + HDIM;
  float* w  = v + HDIM;
  float* dn = w + HDIM;

  const dim3 eb(256), eg((unsigned)(((size_t)HDIM * HDIM) / 256));
  const dim3 gb(256), gg(HDIM / 128, HDIM / 128);

  // X = (delta*A) * 2^-SQUARINGS
  build_x_kernel<<<eg, eb, 0, stream>>>(a_diag, p_vec, q_vec, delta,
                                        1.0f / (float)(1 << SQUARINGS), M0);
  // Horner: R = I + X/N; then R <- I + (X/k) @ R for k = N-1..1
  r_init_kernel<<<eg, eb, 0, stream>>>(M0, M1, 1.0f / (float)TAYLOR_N);
  float* cur = M1;
  float* nxt = M2;
  for (int k = TAYLOR_N - 1; k >= 1; --k) {
    gemm_wmma_f32<false, true, false><<<gg, gb, 0, stream>>>(
        M0, cur, nxt, HDIM, HDIM, HDIM, 1.0f / (float)k);
    float* t = cur; cur = nxt; nxt = t;
  }
  // Squaring: E <- E @ E, SQUARINGS times
  for (int s = 0; s < SQUARINGS; ++s) {
    gemm_wmma_f32<false, false, false><<<gg, gb, 0, stream>>>(
        cur, cur, nxt, HDIM, HDIM, HDIM, 1.0f);
    float* t = cur; cur = nxt; nxt = t;
  }
  float* E  = cur;   // expm(delta*A)
  float* C1 = M0;    // X is dead -> reuse for A^{-1}(E-I)
  float* Bd = nxt;   // dead ping buffer -> B_disc

  // Sherman-Morrison rank-1 inverse applied to (E - I)
  uvd_kernel<<<1, 256, 0, stream>>>(a_diag, p_vec, q_vec, u, v, dn);
  w_kernel<<<HDIM / 256, 256, 0, stream>>>(v, E, w);
  c1_kernel<<<eg, eb, 0, stream>>>(E, a_diag, u, w, dn, C1);

  // B_disc = C1 @ b_mat
  gemm_wmma_f32<false, false, false><<<gg, gb, 0, stream>>>(
      C1, b_mat, Bd, HDIM, HDIM, HDIM, 1.0f);

  // out = h @ E^T ; out += x @ B_disc^T
  gemm_wmma_f32<true, false, false><<<gg, gb, 0, stream>>>(
      h, E, out, HDIM, HDIM, HDIM, 1.0f);
  gemm_wmma_f32<true, false, true><<<gg, gb, 0, stream>>>(
      x, Bd, out, HDIM, HDIM, HDIM, 1.0f);
}